// CenterLoss_19035295056198
// MI455X (gfx1250) — compile-verified
//
#include <hip/hip_runtime.h>

// CenterLoss for MI455X (gfx1250, wave32).
// B = NUM_CLASSES = 16384 samples, D = FEAT_DIM = 1024 features, f32.
// Memory-bound (~128-268 MB traffic vs 134 MFLOP): design centers on b128
// gathered loads + uniform scalar index loads. WMMA f32 16x16x4 (B = ones)
// performs the in-wave partial-sum reduction on the matrix pipe.

#define NUM_CLASSES 16384
#define FEAT_DIM    1024
#define ALPHA       0.5f

typedef __attribute__((ext_vector_type(2))) float v2f;
typedef __attribute__((ext_vector_type(8))) float v8f;

// ---------------------------------------------------------------------------
// Kernel 1: zero the per-class counts (workspace is poisoned by the harness).
// ---------------------------------------------------------------------------
__global__ __launch_bounds__(256) void cl_zero_counts(int* __restrict__ counts) {
    int idx = blockIdx.x * 256 + threadIdx.x;
    if (idx < NUM_CLASSES) counts[idx] = 0;
}

// ---------------------------------------------------------------------------
// Kernel 2: histogram of y_true. Integer atomics -> order-independent,
// deterministic result.
// ---------------------------------------------------------------------------
__global__ __launch_bounds__(256) void cl_hist(const int* __restrict__ y_true,
                                               int* __restrict__ counts) {
    int idx = blockIdx.x * 256 + threadIdx.x;
    if (idx < NUM_CLASSES) {
        atomicAdd(&counts[y_true[idx]], 1);
    }
}

// ---------------------------------------------------------------------------
// Kernel 3: main pass. One 256-thread block per sample row i.
//   j  = y_true[i], jj = y_true[j]            (uniform -> scalar loads)
//   scale = ALPHA / (counts[jj] + 1)          (uniform)
//   diff[d] = y_pred[i,d] - centers[j,d] + scale*(centers[jj,d] - y_pred[j,d])
//   partials[i] = sum_d diff[d]^2
// Each thread handles 4 contiguous floats (float4 / global_load_b128; rows are
// 4KB-strided so every access is 16B aligned).
// Wave reduction via V_WMMA_F32_16X16X4_F32 with B = ones:
//   A[m,0] = p(lane m), A[m,2] = p(lane m+16), A[m,1]=A[m,3]=0
//   => D[m,n] = p(m) + p(m+16)  (independent of n)
//   lane-local sum of the 8 D VGPRs gives the half-wave sum; one shfl_xor(16)
//   + add gives the full wave32 sum in every lane.
// ---------------------------------------------------------------------------
__global__ __launch_bounds__(256) void cl_main(const int*   __restrict__ y_true,
                                               const float* __restrict__ y_pred,
                                               const float* __restrict__ centers,
                                               const int*   __restrict__ counts,
                                               float*       __restrict__ partials) {
    const int i = blockIdx.x;
    const int t = threadIdx.x;

    // Uniform per-block scalars (compiler scalarizes these to SMEM loads).
    const int   j     = y_true[i];
    const int   jj    = y_true[j];
    const float scale = ALPHA / ((float)counts[jj] + 1.0f);

    const int col = t * 4;  // 4 floats per thread, FEAT_DIM = 256*4
    const float* rowA  = y_pred  + (size_t)i  * FEAT_DIM + col;  // y_pred[i]
    const float* rowC1 = centers + (size_t)j  * FEAT_DIM + col;  // centers[j]
    const float* rowC2 = centers + (size_t)jj * FEAT_DIM + col;  // centers[jj]
    const float* rowB  = y_pred  + (size_t)j  * FEAT_DIM + col;  // y_pred[j]

    const float4 a  = *reinterpret_cast<const float4*>(rowA);
    const float4 c1 = *reinterpret_cast<const float4*>(rowC1);
    const float4 c2 = *reinterpret_cast<const float4*>(rowC2);
    const float4 b  = *reinterpret_cast<const float4*>(rowB);

    const float d0 = (a.x - c1.x) + scale * (c2.x - b.x);
    const float d1 = (a.y - c1.y) + scale * (c2.y - b.y);
    const float d2 = (a.z - c1.z) + scale * (c2.z - b.z);
    const float d3 = (a.w - c1.w) + scale * (c2.w - b.w);

    float p = (d0 * d0 + d1 * d1) + (d2 * d2 + d3 * d3);

    // ---- wave32 reduction on the matrix pipe ----
    v2f Am;
    Am[0] = p;      // K=0 (lanes 0-15) / K=2 (lanes 16-31)
    Am[1] = 0.0f;   // K=1 / K=3
    v2f Bm;
    Bm[0] = 1.0f;
    Bm[1] = 1.0f;   // B = all ones (layout-agnostic)
    v8f Cz = {};
    v8f Dm = __builtin_amdgcn_wmma_f32_16x16x4_f32(
        /*neg_a=*/false, Am, /*neg_b=*/false, Bm,
        /*c_mod=*/(short)0, Cz, /*reuse_a=*/false, /*reuse_b=*/false);

    float s = ((Dm[0] + Dm[1]) + (Dm[2] + Dm[3])) +
              ((Dm[4] + Dm[5]) + (Dm[6] + Dm[7]));   // half-wave sum
    s += __shfl_xor(s, 16, 32);                      // full wave sum, all lanes

    // ---- cross-wave reduction (8 waves/block), fixed order ----
    __shared__ float wsum[8];
    const int lane = t & 31;
    const int wave = t >> 5;
    if (lane == 0) wsum[wave] = s;
    __syncthreads();
    if (t == 0) {
        float tot = 0.0f;
#pragma unroll
        for (int w = 0; w < 8; ++w) tot += wsum[w];
        partials[i] = tot;
    }
}

// ---------------------------------------------------------------------------
// Kernel 4: deterministic final reduction of the 16384 block partials.
// Fixed summation order, double accumulation for stability.
// ---------------------------------------------------------------------------
__global__ __launch_bounds__(256) void cl_final(const float* __restrict__ partials,
                                                float* __restrict__ out) {
    __shared__ double buf[256];
    double s = 0.0;
    for (int k = (int)threadIdx.x; k < NUM_CLASSES; k += 256) {
        s += (double)partials[k];
    }
    buf[threadIdx.x] = s;
    __syncthreads();
    if (threadIdx.x == 0) {
        double tot = 0.0;
        for (int k = 0; k < 256; ++k) tot += buf[k];
        out[0] = (float)(tot / ((double)NUM_CLASSES * (double)FEAT_DIM));
    }
}

// ---------------------------------------------------------------------------
// Host-side launcher. Inputs (setup_inputs order): y_true i32[16384],
// y_pred f32[16384*1024], centers f32[16384*1024]. Output: f32[1].
// Workspace layout: [0, 64KB) int counts; [64KB, 128KB) float partials.
// ---------------------------------------------------------------------------
extern "C" void kernel_launch(void* const* d_in, const int* in_sizes, int n_in,
                              void* d_out, int out_size, void* d_ws, size_t ws_size,
                              hipStream_t stream) {
    (void)in_sizes; (void)n_in; (void)out_size; (void)ws_size;

    const int*   y_true  = (const int*)d_in[0];
    const float* y_pred  = (const float*)d_in[1];
    const float* centers = (const float*)d_in[2];
    float*       out     = (float*)d_out;

    int*   counts   = (int*)d_ws;
    float* partials = (float*)((char*)d_ws + (size_t)NUM_CLASSES * sizeof(int));

    const dim3 blk(256);
    cl_zero_counts<<<NUM_CLASSES / 256, blk, 0, stream>>>(counts);
    cl_hist<<<NUM_CLASSES / 256, blk, 0, stream>>>(y_true, counts);
    cl_main<<<NUM_CLASSES, blk, 0, stream>>>(y_true, y_pred, centers, counts, partials);
    cl_final<<<1, blk, 0, stream>>>(partials, out);
}